// DMAQ_QattenMixer_17454747091733
// MI455X (gfx1250) — compile-verified
//
#include <hip/hip_runtime.h>
#include <math.h>

typedef __attribute__((ext_vector_type(2))) float v2f;
typedef __attribute__((ext_vector_type(4))) float v4f;
typedef __attribute__((ext_vector_type(8))) float v8f;

#define N_AGENTS   16
#define ACTION_DIM 256
#define N_KERNEL   4
#define N_HEAD     4
#define EMBED      64
#define LDS_STRIDE 260            // 256 + 4 pad -> conflict-free ds_load_b64
#define WAVES_PER_BLOCK 4
#define ROWS_PER_WAVE   16
#define ROWS_PER_BLOCK  (WAVES_PER_BLOCK * ROWS_PER_WAVE)
#define MAX_BLOCKS      2048      // persistent blocks; >= a few waves per SIMD

// fast sigmoid: v_exp_f32 + v_rcp_f32 (avoids IEEE divide expansion)
__device__ __forceinline__ float sigmoidf_(float x) {
  return __builtin_amdgcn_rcpf(1.0f + __expf(-x));
}

// ---------------------------------------------------------------------------
// Prep: scalar Qatten constants -> out tail; kern[k,a] -> workspace.
// ---------------------------------------------------------------------------
__global__ void dmaq_prep_kernel(const float* __restrict__ selectors,
                                 const float* __restrict__ keys,
                                 const float* __restrict__ si_keys,
                                 const float* __restrict__ si_agents,
                                 float* __restrict__ kern_ws,
                                 float* __restrict__ out_tail) {
  __shared__ float logit_sq[N_HEAD];
  const int t = threadIdx.x;   // 64 threads
  if (t < N_KERNEL * N_AGENTS) {
    const int k = t >> 4, a = t & 15;
    kern_ws[t] = (fabsf(si_keys[k]) + 1e-10f) *
                 sigmoidf_(si_agents[k * N_AGENTS + a]);
  }
  if (t < N_HEAD) {
    float d = 0.0f;
    for (int e = 0; e < EMBED; ++e)
      d = fmaf(selectors[t * EMBED + e], keys[t * EMBED + e], d);
    logit_sq[t] = d * d;
    // softmax over identical logits -> exactly 1/16 each
    const float p = 1.0f / 16.0f;
    out_tail[1 + t] = -16.0f * (logf(p + 1e-8f) * p);   // head entropy
  }
  __syncthreads();
  if (t == 0) {
    out_tail[0] = 0.001f * (logit_sq[0] + logit_sq[1] +
                            logit_sq[2] + logit_sq[3]); // attend_mag_regs
  }
}

// ---------------------------------------------------------------------------
// Main: GEMM acts(B x 256) * si_W^T(256 x 64) via v_wmma_f32_16x16x4_f32,
// sigmoid/kern epilogue, 16-agent shuffle reduction -> v_tot(B).
// Persistent blocks: stage si_W into LDS once, grid-stride over row tiles.
// One wave = 16 batch rows x all 64 output cols (4 N-tiles).
// ---------------------------------------------------------------------------
__global__ void dmaq_main_kernel(const float* __restrict__ agent_qs,
                                 const float* __restrict__ actions,
                                 const float* __restrict__ max_q_i,
                                 const float* __restrict__ si_W,
                                 const float* __restrict__ si_b,
                                 const float* __restrict__ kern_ws,
                                 float* __restrict__ v_tot,
                                 int nrows) {
  extern __shared__ float ldsW[];   // 64 rows x LDS_STRIDE floats

  const int tid = threadIdx.x;

  // Stage si_W (64 x 256 f32) into padded LDS once, coalesced b128 loads.
  for (int i = tid * 4; i < N_KERNEL * N_AGENTS * ACTION_DIM;
       i += WAVES_PER_BLOCK * 32 * 4) {
    const int r = i >> 8;
    const int c = i & 255;
    v4f w = *(const v4f*)(si_W + (long)r * ACTION_DIM + c);
    *(v4f*)(ldsW + r * LDS_STRIDE + c) = w;   // 16B aligned: 260*4 % 16 == 0
  }
  __syncthreads();

  const int lane = tid & 31;
  const int wave = tid >> 5;
  const int a16  = lane & 15;      // agent / N index, and A-matrix row M
  const int hi   = lane >> 4;      // K half-select for A/B fragments

  // Per-lane epilogue constants (agent = a16, N-tile t = SI kernel index).
  const float kn0 = kern_ws[0 * N_AGENTS + a16];
  const float kn1 = kern_ws[1 * N_AGENTS + a16];
  const float kn2 = kern_ws[2 * N_AGENTS + a16];
  const float kn3 = kern_ws[3 * N_AGENTS + a16];
  const float bb0 = si_b[0 * N_AGENTS + a16];
  const float bb1 = si_b[1 * N_AGENTS + a16];
  const float bb2 = si_b[2 * N_AGENTS + a16];
  const float bb3 = si_b[3 * N_AGENTS + a16];
  const float wf  = 4.0f * (1.0f / 16.0f) + 1e-10f;   // w_final (v cancels)

  // B fragments (4x16 f32 per N-tile): lane holds col N=a16, K = k0+2*hi+{0,1}
  const float* L0 = ldsW + (0 * N_AGENTS + a16) * LDS_STRIDE + 2 * hi;
  const float* L1 = L0 + N_AGENTS * LDS_STRIDE;
  const float* L2 = L1 + N_AGENTS * LDS_STRIDE;
  const float* L3 = L2 + N_AGENTS * LDS_STRIDE;

  const int ntiles = (nrows + ROWS_PER_BLOCK - 1) / ROWS_PER_BLOCK;
  for (int tile = blockIdx.x; tile < ntiles; tile += gridDim.x) {
    const long rowBase = ((long)tile * WAVES_PER_BLOCK + wave) * ROWS_PER_WAVE;
    if (rowBase >= nrows) continue;   // wave-uniform: EXEC stays all-ones

    // A fragment (16x4 f32): lane holds row M=a16, K = k0 + 2*hi + {0,1}
    const float* Arow = actions + (rowBase + a16) * (long)ACTION_DIM + 2 * hi;

    v8f c0 = {}, c1 = {}, c2 = {}, c3 = {};
#pragma unroll 4
    for (int k0 = 0; k0 < ACTION_DIM; k0 += 4) {
      v2f a  = *(const v2f*)(Arow + k0);
      v2f b0 = *(const v2f*)(L0 + k0);
      v2f b1 = *(const v2f*)(L1 + k0);
      v2f b2 = *(const v2f*)(L2 + k0);
      v2f b3 = *(const v2f*)(L3 + k0);
      c0 = __builtin_amdgcn_wmma_f32_16x16x4_f32(false, a, false, b0, (short)0, c0, false, false);
      c1 = __builtin_amdgcn_wmma_f32_16x16x4_f32(false, a, false, b1, (short)0, c1, false, false);
      c2 = __builtin_amdgcn_wmma_f32_16x16x4_f32(false, a, false, b2, (short)0, c2, false, false);
      c3 = __builtin_amdgcn_wmma_f32_16x16x4_f32(false, a, false, b3, (short)0, c3, false, false);
    }

#pragma unroll
    for (int r = 0; r < 8; ++r) {
      // C/D layout: VGPR r holds M = r + 8*hi, N = a16
      const float advw = sigmoidf_(c0[r] + bb0) * kn0
                       + sigmoidf_(c1[r] + bb1) * kn1
                       + sigmoidf_(c2[r] + bb2) * kn2
                       + sigmoidf_(c3[r] + bb3) * kn3;
      const long row = rowBase + r + 8 * hi;
      const float dq = agent_qs[row * N_AGENTS + a16] - max_q_i[row * N_AGENTS + a16];
      float contrib = wf * dq * (advw - 1.0f);
      // reduce over the 16 agent lanes (stays within each half of the wave)
      contrib += __shfl_xor(contrib, 1, 32);
      contrib += __shfl_xor(contrib, 2, 32);
      contrib += __shfl_xor(contrib, 4, 32);
      contrib += __shfl_xor(contrib, 8, 32);
      if (a16 == 0) v_tot[row] = contrib;
    }
  }
}

// ---------------------------------------------------------------------------
extern "C" void kernel_launch(void* const* d_in, const int* in_sizes, int n_in,
                              void* d_out, int out_size, void* d_ws, size_t ws_size,
                              hipStream_t stream) {
  (void)n_in; (void)out_size; (void)ws_size;
  const float* agent_qs  = (const float*)d_in[0];
  const float* actions   = (const float*)d_in[1];
  const float* max_q_i   = (const float*)d_in[2];
  const float* selectors = (const float*)d_in[3];
  const float* keysp     = (const float*)d_in[4];
  /* d_in[5] = V: cancels in adv_q, unused */
  const float* si_keys   = (const float*)d_in[6];
  const float* si_agents = (const float*)d_in[7];
  const float* si_W      = (const float*)d_in[8];
  const float* si_b      = (const float*)d_in[9];

  float* out      = (float*)d_out;
  const int nrows = in_sizes[0] / N_AGENTS;     // B0*T
  float* kern_ws  = (float*)d_ws;               // 64 floats
  float* out_tail = out + nrows;                // [regs, ent0..ent3]

  dmaq_prep_kernel<<<1, 64, 0, stream>>>(selectors, keysp, si_keys, si_agents,
                                         kern_ws, out_tail);

  const int ntiles  = (nrows + ROWS_PER_BLOCK - 1) / ROWS_PER_BLOCK;
  const int nblocks = ntiles < MAX_BLOCKS ? ntiles : MAX_BLOCKS;
  const size_t ldsBytes = (size_t)(N_KERNEL * N_AGENTS) * LDS_STRIDE * sizeof(float);
  dmaq_main_kernel<<<nblocks, WAVES_PER_BLOCK * 32, ldsBytes, stream>>>(
      agent_qs, actions, max_q_i, si_W, si_b, kern_ws, out, nrows);
}